// Decode_10728828305870
// MI455X (gfx1250) — compile-verified
//
#include <hip/hip_runtime.h>
#include <math.h>

// Problem constants (from reference)
#define NP 64
#define NR 8
#define NK 500
#define HID 256
#define HEADS 8
#define VD 8
#define HD 32           // HID / HEADS
#define BATCH (NP * NR) // 512
#define G4 (4 * HID)    // 1024

typedef __attribute__((ext_vector_type(2))) float v2f;
typedef __attribute__((ext_vector_type(8))) float v8f;

__device__ __forceinline__ float sigmoidf(float x) {
    return 1.0f / (1.0f + __expf(-x));
}

// CDNA5 hardware tanh (trans op). ISA 7.4: one independent op / V_NOP after a
// trans op before its output is consumed -> bundle a v_nop for safety.
__device__ __forceinline__ float fast_tanh(float x) {
    float r;
    asm("v_tanh_f32 %0, %1\n\tv_nop" : "=v"(r) : "v"(x));
    return r;
}

// ---------------------------------------------------------------------------
// K1: gates[512,1024] = query@W_ih^T + state1@W_hh^T + b_ih + b_hh
// One 16x16 output tile per wave, f32 WMMA, K=256 in steps of 4.
// ---------------------------------------------------------------------------
__global__ __launch_bounds__(128) void lstm_gates_wmma(
    const float* __restrict__ query, const float* __restrict__ state1,
    const float* __restrict__ W_ih, const float* __restrict__ W_hh,
    const float* __restrict__ b_ih, const float* __restrict__ b_hh,
    float* __restrict__ gates)
{
    const int wave  = threadIdx.x >> 5;
    const int lane  = threadIdx.x & 31;
    const int tile  = blockIdx.x * 4 + wave;      // 2048 tiles
    const int tileM = (tile >> 6) << 4;           // 32 tiles of M
    const int tileN = (tile & 63) << 4;           // 64 tiles of N
    const int m     = lane & 15;
    const int khalf = (lane >> 4) << 1;           // 0 or 2

    const float* aq = query  + (tileM + m) * HID;
    const float* as = state1 + (tileM + m) * HID;
    const float* bq = W_ih   + (tileN + m) * HID; // W row = output col
    const float* bs = W_hh   + (tileN + m) * HID;

    v8f acc = {};
    for (int kk = 0; kk < HID; kk += 4) {
        const int k0 = kk + khalf;
        v2f a1 = *(const v2f*)(aq + k0);
        v2f b1 = *(const v2f*)(bq + k0);
        v2f a2 = *(const v2f*)(as + k0);
        v2f b2 = *(const v2f*)(bs + k0);
        acc = __builtin_amdgcn_wmma_f32_16x16x4_f32(false, a1, false, b1,
                                                    (short)0, acc, false, false);
        acc = __builtin_amdgcn_wmma_f32_16x16x4_f32(false, a2, false, b2,
                                                    (short)0, acc, false, false);
    }
    // C/D layout: VGPR v -> row v (lanes 0-15) / row v+8 (lanes 16-31), col = l%16
    const int n     = tileN + m;
    const float bias = b_ih[n] + b_hh[n];
    const int rbase = tileM + ((lane >> 4) << 3);
#pragma unroll
    for (int v = 0; v < 8; ++v)
        gates[(rbase + v) * G4 + n] = acc[v] + bias;
}

// ---------------------------------------------------------------------------
// K2: LSTM gate activations -> h, c  (gate order i, f, g, o).  Precise tanhf:
// h/c are direct outputs, cost is negligible (131K elements).
// ---------------------------------------------------------------------------
__global__ __launch_bounds__(256) void lstm_act(
    const float* __restrict__ gates, const float* __restrict__ state2,
    float* __restrict__ h_out, float* __restrict__ c_out)
{
    const int idx = blockIdx.x * 256 + threadIdx.x;   // BATCH*HID
    const int b  = idx >> 8;
    const int hc = idx & 255;
    const float gi = gates[b * G4 + hc];
    const float gf = gates[b * G4 + HID + hc];
    const float gg = gates[b * G4 + 2 * HID + hc];
    const float go = gates[b * G4 + 3 * HID + hc];
    const float c  = sigmoidf(gf) * state2[idx] + sigmoidf(gi) * tanhf(gg);
    c_out[idx] = c;
    h_out[idx] = sigmoidf(go) * tanhf(c);
}

// ---------------------------------------------------------------------------
// K3: glimpse multi-head attention; one block per (head, problem).
// ---------------------------------------------------------------------------
__global__ __launch_bounds__(256) void attention_kernel(
    const float* __restrict__ h,    // [BATCH,HID]
    const float* __restrict__ Kt,   // [HEADS,NP,NK,HD]
    const float* __restrict__ Vt,   // [HEADS,NP,NK,HD]
    const float* __restrict__ nnQ,  // [HEADS,HID,HD]
    const int*   __restrict__ mask, // [NP,NR,NK]
    float* __restrict__ xout)       // [BATCH,HID] (head-major inner HD)
{
    __shared__ float q_s[NR * HID];   // 8 KB
    __shared__ float Q_s[NR * HD];    // 1 KB
    __shared__ float S_s[NR * NK];    // ~16 KB
    const int a   = blockIdx.x & (HEADS - 1);
    const int p   = blockIdx.x >> 3;
    const int tid = threadIdx.x;

    for (int idx = tid; idx < NR * HID; idx += 256)
        q_s[idx] = h[p * NR * HID + idx];
    __syncthreads();

    {   // Q[r][d] = sum_h q[r][h] * nnQ[a][h][d], 1/sqrt(HD) folded in
        const int r = tid >> 5, d = tid & 31;
        const float* w = nnQ + (a * HID) * HD + d;
        float s = 0.f;
        for (int hh = 0; hh < HID; ++hh) s += q_s[r * HID + hh] * w[hh * HD];
        Q_s[r * HD + d] = s * 0.17677669529663687f;
    }
    __syncthreads();

    for (int idx = tid; idx < NR * NK; idx += 256) {
        const int r = idx / NK, k = idx - r * NK;
        const float* kv = Kt + (((size_t)a * NP + p) * NK + k) * HD;
        float s = 0.f;
#pragma unroll
        for (int d = 0; d < HD; ++d) s += Q_s[r * HD + d] * kv[d];
        if (mask[(p * NR + r) * NK + k]) s = -INFINITY;
        S_s[idx] = s;
    }
    __syncthreads();

    {   // softmax: wave w handles row r = w
        const int r = tid >> 5, lane = tid & 31;
        float mx = -INFINITY;
        for (int k = lane; k < NK; k += 32) mx = fmaxf(mx, S_s[r * NK + k]);
#pragma unroll
        for (int off = 16; off; off >>= 1) mx = fmaxf(mx, __shfl_xor(mx, off, 32));
        float sum = 0.f;
        for (int k = lane; k < NK; k += 32) {
            const float e = __expf(S_s[r * NK + k] - mx);
            S_s[r * NK + k] = e;
            sum += e;
        }
#pragma unroll
        for (int off = 16; off; off >>= 1) sum += __shfl_xor(sum, off, 32);
        const float inv = 1.0f / sum;
        for (int k = lane; k < NK; k += 32) S_s[r * NK + k] *= inv;
    }
    __syncthreads();

    {   // x[r][d] = sum_k S[r][k] * V[k][d]
        const int r = tid >> 5, d = tid & 31;
        const float* vv = Vt + (((size_t)a * NP + p) * NK) * HD + d;
        float s = 0.f;
        for (int k = 0; k < NK; ++k) s += S_s[r * NK + k] * vv[k * HD];
        xout[(p * NR + r) * HID + a * HD + d] = s;
    }
}

// ---------------------------------------------------------------------------
// K4: qm[512,256] = x[512,256] @ nn_O[256,256]  (B stored k-major)
// ---------------------------------------------------------------------------
__global__ __launch_bounds__(128) void gemm_xO_wmma(
    const float* __restrict__ A, const float* __restrict__ W,
    float* __restrict__ out)
{
    const int wave  = threadIdx.x >> 5;
    const int lane  = threadIdx.x & 31;
    const int tile  = blockIdx.x * 4 + wave;      // 512 tiles
    const int tileM = (tile >> 4) << 4;           // 32 M-tiles
    const int tileN = (tile & 15) << 4;           // 16 N-tiles
    const int m     = lane & 15;
    const int khalf = (lane >> 4) << 1;
    const float* ar = A + (tileM + m) * HID;
    const int n     = tileN + m;

    v8f acc = {};
    for (int kk = 0; kk < HID; kk += 4) {
        const int k0 = kk + khalf;
        v2f a = *(const v2f*)(ar + k0);
        v2f b;
        b[0] = W[k0 * HID + n];
        b[1] = W[(k0 + 1) * HID + n];
        acc = __builtin_amdgcn_wmma_f32_16x16x4_f32(false, a, false, b,
                                                    (short)0, acc, false, false);
    }
    const int rbase = tileM + ((lane >> 4) << 3);
#pragma unroll
    for (int v = 0; v < 8; ++v)
        out[(rbase + v) * HID + n] = acc[v];
}

// ---------------------------------------------------------------------------
// K5: fused additive-attention scoring + argmax + log-softmax per (p, r).
// X rows stream through LDS via double-buffered global_load_async_to_lds_b128
// (ASYNCcnt-tracked, overlaps fetch with the tanh.w reduction); vf recomputed
// on the fly; hardware v_tanh_f32 in the hot loop.
// ---------------------------------------------------------------------------
__global__ __launch_bounds__(256) void logits_choose(
    const float* __restrict__ X,        // [NP,NK,HID]
    const float* __restrict__ varfeat,  // [NP,VD,NK]
    const float* __restrict__ qm,       // [BATCH,HID]
    const float* __restrict__ nnA,      // [VD,HID]
    const float* __restrict__ nnB,      // [HID]
    const float* __restrict__ nnW,      // [HID]
    const int*   __restrict__ mask,     // [BATCH,NK]
    float* __restrict__ chosen_p)       // [BATCH]
{
    __shared__ float xstage[2][8][HID]; // 16 KB double-buffered X rows
    __shared__ float l_s[NK];
    __shared__ float redv[256];
    __shared__ int   redi[256];
    const int b    = blockIdx.x;        // BATCH
    const int p    = b >> 3;            // / NR
    const int wave = threadIdx.x >> 5;
    const int lane = threadIdx.x & 31;

    // per-lane loop-invariants for h_j = lane + 32*j
    float wj[8], qb[8], Aj[8][VD];
#pragma unroll
    for (int j = 0; j < 8; ++j) {
        const int hh = lane + 32 * j;
        wj[j] = nnW[hh];
        qb[j] = qm[b * HID + hh] + nnB[hh];
#pragma unroll
        for (int v = 0; v < VD; ++v) Aj[j][v] = nnA[v * HID + hh];
    }

    // async issue: each lane moves 16B; two issues (offset 0 / 512) per 1KB row
    const unsigned lds0 = (unsigned)(size_t)(&xstage[0][wave][0]) + lane * 16u;
    const unsigned lds1 = (unsigned)(size_t)(&xstage[1][wave][0]) + lane * 16u;
#define ISSUE_ROW(buf, k)                                                      \
    do {                                                                       \
        const float* g_ = X + ((size_t)p * NK + (k)) * HID + lane * 4;         \
        unsigned l_ = (buf) ? lds1 : lds0;                                     \
        asm volatile("global_load_async_to_lds_b128 %0, %1, off"               \
                     :: "v"(l_), "v"(g_) : "memory");                          \
        asm volatile("global_load_async_to_lds_b128 %0, %1, off offset:512"    \
                     :: "v"(l_), "v"(g_) : "memory");                          \
    } while (0)

    ISSUE_ROW(0, wave);
    int buf = 0;
    for (int k = wave; k < NK; k += 8) {
        const int kn = k + 8;
        if (kn < NK) {
            ISSUE_ROW(buf ^ 1, kn);
            asm volatile("s_wait_asynccnt 0x2" ::: "memory"); // current row done
        } else {
            asm volatile("s_wait_asynccnt 0x0" ::: "memory");
        }
        float vfv[VD];
#pragma unroll
        for (int v = 0; v < VD; ++v) vfv[v] = varfeat[(p * VD + v) * NK + k];
        const float* xr = &xstage[buf][wave][0];
        float acc = 0.f;
#pragma unroll
        for (int j = 0; j < 8; ++j) {
            float t = xr[lane + 32 * j] + qb[j];
#pragma unroll
            for (int v = 0; v < VD; ++v) t = fmaf(vfv[v], Aj[j][v], t);
            acc += fast_tanh(t) * wj[j];
        }
#pragma unroll
        for (int off = 16; off; off >>= 1) acc += __shfl_xor(acc, off, 32);
        if (lane == 0)
            l_s[k] = mask[b * NK + k] ? -INFINITY : fast_tanh(acc) * 10.0f;
        buf ^= 1;
    }
#undef ISSUE_ROW
    __syncthreads();

    // block argmax (first occurrence on ties)
    float bm = -INFINITY; int bi = 0x7fffffff;
    for (int k = threadIdx.x; k < NK; k += 256)
        if (l_s[k] > bm) { bm = l_s[k]; bi = k; }
    redv[threadIdx.x] = bm; redi[threadIdx.x] = bi;
    __syncthreads();
    for (int s = 128; s; s >>= 1) {
        if (threadIdx.x < (unsigned)s) {
            const float ov = redv[threadIdx.x + s];
            const int   oi = redi[threadIdx.x + s];
            if (ov > redv[threadIdx.x] ||
                (ov == redv[threadIdx.x] && oi < redi[threadIdx.x])) {
                redv[threadIdx.x] = ov; redi[threadIdx.x] = oi;
            }
        }
        __syncthreads();
    }
    const float maxv  = redv[0];
    const int   besti = redi[0];
    __syncthreads();

    float psum = 0.f;
    for (int k = threadIdx.x; k < NK; k += 256) psum += __expf(l_s[k] - maxv);
    redv[threadIdx.x] = psum;
    __syncthreads();
    for (int s = 128; s; s >>= 1) {
        if (threadIdx.x < (unsigned)s) redv[threadIdx.x] += redv[threadIdx.x + s];
        __syncthreads();
    }
    if (threadIdx.x == 0)
        chosen_p[b] = l_s[besti] - maxv - __logf(redv[0]);
}

// ---------------------------------------------------------------------------
extern "C" void kernel_launch(void* const* d_in, const int* in_sizes, int n_in,
                              void* d_out, int out_size, void* d_ws, size_t ws_size,
                              hipStream_t stream) {
    const float* X       = (const float*)d_in[0];
    const float* Kt      = (const float*)d_in[1];
    const float* Vt      = (const float*)d_in[2];
    const float* query   = (const float*)d_in[3];
    const float* state1  = (const float*)d_in[4];
    const float* state2  = (const float*)d_in[5];
    const float* varfeat = (const float*)d_in[6];
    const int*   mask    = (const int*)d_in[7];
    const float* nnQ     = (const float*)d_in[8];
    const float* nnO     = (const float*)d_in[9];
    const float* nnA     = (const float*)d_in[10];
    const float* nnB     = (const float*)d_in[11];
    const float* nnW     = (const float*)d_in[12];
    const float* W_ih    = (const float*)d_in[13];
    const float* W_hh    = (const float*)d_in[14];
    const float* b_ih    = (const float*)d_in[15];
    const float* b_hh    = (const float*)d_in[16];

    float* h_out  = (float*)d_out;                      // [512,256]
    float* c_out  = h_out + BATCH * HID;                // [512,256]
    float* cp_out = c_out + BATCH * HID;                // [512]

    float* ws    = (float*)d_ws;
    float* gates = ws;                                  // 512*1024
    float* x_att = gates + BATCH * G4;                  // 512*256
    float* qmbuf = x_att + BATCH * HID;                 // 512*256

    lstm_gates_wmma<<<512, 128, 0, stream>>>(query, state1, W_ih, W_hh,
                                             b_ih, b_hh, gates);
    lstm_act<<<(BATCH * HID) / 256, 256, 0, stream>>>(gates, state2, h_out, c_out);
    attention_kernel<<<HEADS * NP, 256, 0, stream>>>(h_out, Kt, Vt, nnQ, mask, x_att);
    gemm_xO_wmma<<<128, 128, 0, stream>>>(x_att, nnO, qmbuf);
    logits_choose<<<BATCH, 256, 0, stream>>>(X, varfeat, qmbuf, nnA, nnB, nnW,
                                             mask, cp_out);
}